// VQC_2954937500421
// MI455X (gfx1250) — compile-verified
//
#include <hip/hip_runtime.h>

// ---------------------------------------------------------------------------
// 12-qubit VQC simulator, one sample per workgroup, state resident in LDS.
// Single-qubit gate sweeps fused into 16x16 complex unitaries applied with
// V_WMMA_F32_16X16X4_F32 (f32 matches the complex64 reference math).
// ---------------------------------------------------------------------------

typedef float v2f __attribute__((ext_vector_type(2)));
typedef float v8f __attribute__((ext_vector_type(8)));

#define NWIRES 12
#define NLAYERS 4
#define NOUT 4
#define SDIM 4096
#define NTHREADS 256

__device__ __forceinline__ v8f wmma4(v2f a, v2f b, v8f c) {
  // D = A(16x4) * B(4x16) + C, all f32
  return __builtin_amdgcn_wmma_f32_16x16x4_f32(
      /*neg_a=*/false, a, /*neg_b=*/false, b,
      /*c_mod=*/(short)0, c, /*reuse_a=*/false, /*reuse_b=*/false);
}

// Apply the fused 16x16 complex unitary (ur/ui) to the state for wire-group
// `mode` (0: wires 0-3, 1: wires 4-7, 2: wires 8-11). 8 waves x 2 tiles.
__device__ __forceinline__ void apply_group(int mode,
                                            float* __restrict__ sr,
                                            float* __restrict__ si,
                                            const float* __restrict__ ur,
                                            const float* __restrict__ ui) {
  const int tid = threadIdx.x;
  const int wv  = tid >> 5;   // wave id 0..7
  const int l   = tid & 31;   // lane id
  const int h   = l >> 4;     // lane half
  const int lm  = l & 15;

  for (int t = wv * 2; t < wv * 2 + 2; ++t) {
    v8f ar = {};
    v8f ai = {};
    for (int k = 0; k < 4; ++k) {
      const int K0 = k * 4 + 2 * h;  // K for component .x ; .y is K0+1

      // U fragment: rows m=lm (A layout) OR U^T as B with n=lm -- same index.
      v2f fUr, fUi;
      fUr.x = ur[lm * 16 + K0];  fUr.y = ur[lm * 16 + K0 + 1];
      fUi.x = ui[lm * 16 + K0];  fUi.y = ui[lm * 16 + K0 + 1];

      // State fragment (B operand for modes 0/1, A operand for mode 2).
      int x0, x1;
      if (mode == 0)      { x0 = K0 * 256 + t * 16 + lm;  x1 = x0 + 256; }
      else if (mode == 1) { x0 = t * 256 + K0 * 16 + lm;  x1 = x0 + 16;  }
      else                { x0 = t * 256 + lm * 16 + K0;  x1 = x0 + 1;   }
      v2f fXr, fXi;
      fXr.x = sr[x0];  fXr.y = sr[x1];
      fXi.x = si[x0];  fXi.y = si[x1];

      if (mode != 2) {
        // D = U4 * X  (complex): Dr = Ur*Xr - Ui*Xi ; Di = Ur*Xi + Ui*Xr
        ar = wmma4(fUr, fXr, ar);
        ar = wmma4(-fUi, fXi, ar);
        ai = wmma4(fUr, fXi, ai);
        ai = wmma4(fUi, fXr, ai);
      } else {
        // D = X * U4^T (complex): Dr = Xr*Ur - Xi*Ui ; Di = Xr*Ui + Xi*Ur
        ar = wmma4(fXr, fUr, ar);
        ar = wmma4(-fXi, fUi, ar);
        ai = wmma4(fXr, fUi, ai);
        ai = wmma4(fXi, fUr, ai);
      }
    }
    // Write D back (in-place is safe: this tile was fully read above and
    // tiles are disjoint across waves).
    for (int r = 0; r < 8; ++r) {
      const int m = r + 8 * h;
      int o;
      if (mode == 0) o = m * 256 + t * 16 + lm;
      else           o = t * 256 + m * 16 + lm;  // modes 1 and 2 coincide
      sr[o] = ar[r];
      si[o] = ai[r];
    }
  }
}

// Build U4 = g0 (x) g1 (x) g2 (x) g3 from the 4 complex 2x2 gates in gr/gi.
// 256 threads: one U4 entry each.
__device__ __forceinline__ void build_u4(const float* __restrict__ gr,
                                         const float* __restrict__ gi,
                                         float* __restrict__ ur,
                                         float* __restrict__ ui) {
  const int e = threadIdx.x;
  const int p = e >> 4, q = e & 15;
  float pr = 1.0f, pim = 0.0f;
  for (int k = 0; k < 4; ++k) {
    const int pk = (p >> (3 - k)) & 1;
    const int qk = (q >> (3 - k)) & 1;
    const float grk = gr[k * 4 + pk * 2 + qk];
    const float gik = gi[k * 4 + pk * 2 + qk];
    const float nr = pr * grk - pim * gik;
    const float ni = pr * gik + pim * grk;
    pr = nr; pim = ni;
  }
  ur[e] = pr;
  ui[e] = pim;
}

// CNOT(control bit cpos, target bit tpos): in-place swap of the control=1,
// target=0/1 amplitude pairs. 1024 disjoint pairs, 4 per thread.
__device__ __forceinline__ void cnot_pass(float* __restrict__ sr,
                                          float* __restrict__ si,
                                          int cpos, int tpos) {
  for (int idx = threadIdx.x; idx < 1024; idx += NTHREADS) {
    int s = 0, src = idx;
    for (int bpos = 0; bpos < NWIRES; ++bpos) {
      if (bpos == cpos || bpos == tpos) continue;
      s |= (src & 1) << bpos;
      src >>= 1;
    }
    s |= (1 << cpos);            // control = 1
    const int s2 = s | (1 << tpos);  // target partner
    const float tr = sr[s], ti = si[s];
    sr[s] = sr[s2];  si[s] = si[s2];
    sr[s2] = tr;     si[s2] = ti;
  }
}

__global__ __launch_bounds__(NTHREADS) void vqc_kernel(
    const float* __restrict__ in,     // (1024, 12)
    const float* __restrict__ wts,    // (4, 12, 3)
    float* __restrict__ out) {        // (1024, 4)
  __shared__ float sr[SDIM];
  __shared__ float si[SDIM];
  __shared__ float ur[256];
  __shared__ float ui[256];
  __shared__ float gr[16];
  __shared__ float gi[16];
  __shared__ float oacc[NOUT];

  const int b   = blockIdx.x;
  const int tid = threadIdx.x;

  // |0...0> state
  for (int s = tid; s < SDIM; s += NTHREADS) {
    sr[s] = (s == 0) ? 1.0f : 0.0f;
    si[s] = 0.0f;
  }
  if (tid < NOUT) oacc[tid] = 0.0f;

  // ---- AngleEmbedding: RY(pi/2 * x_w), fused 4 wires at a time -----------
  for (int g = 0; g < 3; ++g) {
    __syncthreads();
    if (tid < 4) {
      const int wi = 4 * g + tid;
      const float half = 0.25f * 3.14159265358979323846f * in[b * NWIRES + wi];
      const float c = __builtin_cosf(half);
      const float s = __builtin_sinf(half);
      gr[tid * 4 + 0] = c;   gi[tid * 4 + 0] = 0.0f;
      gr[tid * 4 + 1] = -s;  gi[tid * 4 + 1] = 0.0f;
      gr[tid * 4 + 2] = s;   gi[tid * 4 + 2] = 0.0f;
      gr[tid * 4 + 3] = c;   gi[tid * 4 + 3] = 0.0f;
    }
    __syncthreads();
    build_u4(gr, gi, ur, ui);
    __syncthreads();
    apply_group(g, sr, si, ur, ui);
  }

  // ---- StronglyEntanglingLayers ------------------------------------------
  for (int l = 0; l < NLAYERS; ++l) {
    for (int g = 0; g < 3; ++g) {
      __syncthreads();
      if (tid < 4) {
        const int wi = 4 * g + tid;
        const float* W = wts + (l * NWIRES + wi) * 3;
        const float phi = W[0], th = W[1], om = W[2];
        const float c  = __builtin_cosf(0.5f * th);
        const float s  = __builtin_sinf(0.5f * th);
        const float ap = 0.5f * (phi + om);
        const float am = 0.5f * (phi - om);
        const float ca = __builtin_cosf(ap), sa = __builtin_sinf(ap);
        const float cm = __builtin_cosf(am), sm = __builtin_sinf(am);
        // Rot = RZ(om) RY(th) RZ(phi)
        gr[tid * 4 + 0] =  ca * c;  gi[tid * 4 + 0] = -sa * c;
        gr[tid * 4 + 1] = -cm * s;  gi[tid * 4 + 1] = -sm * s;
        gr[tid * 4 + 2] =  cm * s;  gi[tid * 4 + 2] = -sm * s;
        gr[tid * 4 + 3] =  ca * c;  gi[tid * 4 + 3] =  sa * c;
      }
      __syncthreads();
      build_u4(gr, gi, ur, ui);
      __syncthreads();
      apply_group(g, sr, si, ur, ui);
    }
    // Ring of CNOTs, range r = (l % 11) + 1, applied sequentially.
    const int r = (l % (NWIRES - 1)) + 1;
    for (int wq = 0; wq < NWIRES; ++wq) {
      __syncthreads();
      const int c = wq;
      const int t = (wq + r) % NWIRES;
      cnot_pass(sr, si, (NWIRES - 1) - c, (NWIRES - 1) - t);
    }
  }

  __syncthreads();

  // ---- expval(PauliZ) on wires 0..3 --------------------------------------
  float a0 = 0.f, a1 = 0.f, a2 = 0.f, a3 = 0.f;
  for (int s = tid; s < SDIM; s += NTHREADS) {
    const float p = sr[s] * sr[s] + si[s] * si[s];
    a0 += ((s >> 11) & 1) ? -p : p;
    a1 += ((s >> 10) & 1) ? -p : p;
    a2 += ((s >> 9) & 1) ? -p : p;
    a3 += ((s >> 8) & 1) ? -p : p;
  }
  atomicAdd(&oacc[0], a0);
  atomicAdd(&oacc[1], a1);
  atomicAdd(&oacc[2], a2);
  atomicAdd(&oacc[3], a3);
  __syncthreads();
  if (tid < NOUT) out[b * NOUT + tid] = oacc[tid];
}

extern "C" void kernel_launch(void* const* d_in, const int* in_sizes, int n_in,
                              void* d_out, int out_size, void* d_ws, size_t ws_size,
                              hipStream_t stream) {
  (void)in_sizes; (void)n_in; (void)out_size; (void)d_ws; (void)ws_size;
  const float* inputs  = (const float*)d_in[0];   // (1024, 12) f32
  const float* weights = (const float*)d_in[1];   // (4, 12, 3) f32
  float* out = (float*)d_out;                     // (1024, 4) f32
  vqc_kernel<<<1024, NTHREADS, 0, stream>>>(inputs, weights, out);
}